// GravNetBlock_9225589752202
// MI455X (gfx1250) — compile-verified
//
#include <hip/hip_runtime.h>
#include <cfloat>
#include <cmath>

// Problem constants (match the JAX reference)
#define Bv   16
#define Mv   2048
#define Kv   40
#define Nv   (Bv * Mv)      // 32768
#define INv  96
#define SDv  4
#define PDv  22
#define OUTv 96

typedef __attribute__((ext_vector_type(2))) float v2f;
typedef __attribute__((ext_vector_type(8))) float v8f;

// ---------------------------------------------------------------------------
// Generic fused GEMM on the WMMA pipe:  D = act( A1@W1 [+ A2@W2] + bias )
//   - one wave computes one 16x16 output tile via V_WMMA_F32_16X16X4_F32
//   - K1/K2/P compile-time -> fully unrolled, immediate-offset loads
//   - BIAS_MODE: 0 = none, 1 = bias[col], 2 = bias[(row/2048)*P + col]
//   - ACT: 0 = identity, 1 = tanh
// Ragged P (4, 22): B-column reads are clamped in-bounds once; the garbage
// columns accumulate into lanes whose results are never stored.
// A fragment (ISA 7.12.2, 32-bit A 16x4): lanes 0-15 rows / K=k0+{0,1};
// lanes 16-31 same rows / K=k0+{2,3}. B mirrors. C/D: 8 VGPRs, rows r / r+8.
// ---------------------------------------------------------------------------
template<int K1, int K2, int P, int BIAS_MODE, int ACT>
__global__ void __launch_bounds__(256) gemm_wmma_f32(
    const float* __restrict__ A1, const float* __restrict__ W1,
    const float* __restrict__ A2, const float* __restrict__ W2,
    const float* __restrict__ bias, float* __restrict__ D, int nrows)
{
    constexpr int tilesX = (P + 15) >> 4;
    const int lane  = threadIdx.x & 31;
    const int wave  = (blockIdx.x * blockDim.x + threadIdx.x) >> 5;
    const int tiles = (nrows >> 4) * tilesX;
    if (wave >= tiles) return;

    const int tr    = wave / tilesX;
    const int tc    = wave - tr * tilesX;
    const int row0  = tr << 4;
    const int col0  = tc << 4;
    const int rsel  = lane & 15;          // row (A) / col (B) within tile
    const int khalf = (lane >> 4) << 1;   // 0 or 2
    const int col   = col0 + rsel;
    const int colc  = (col < P) ? col : (P - 1);   // clamp once; loop is guard-free

    v8f c = {0.f, 0.f, 0.f, 0.f, 0.f, 0.f, 0.f, 0.f};

    {
        const float* ap = A1 + (size_t)(row0 + rsel) * K1 + khalf;
        const float* wp = W1 + (size_t)khalf * P + colc;
#pragma unroll
        for (int k0 = 0; k0 < K1; k0 += 4) {
            v2f a, b;
            a.x = ap[k0];
            a.y = ap[k0 + 1];
            b.x = wp[(size_t)k0 * P];
            b.y = wp[(size_t)(k0 + 1) * P];
            c = __builtin_amdgcn_wmma_f32_16x16x4_f32(false, a, false, b, (short)0, c,
                                                      false, false);
        }
    }
    if constexpr (K2 > 0) {
        const float* ap = A2 + (size_t)(row0 + rsel) * K2 + khalf;
        const float* wp = W2 + (size_t)khalf * P + colc;
#pragma unroll
        for (int k0 = 0; k0 < K2; k0 += 4) {
            v2f a, b;
            a.x = ap[k0];
            a.y = ap[k0 + 1];
            b.x = wp[(size_t)k0 * P];
            b.y = wp[(size_t)(k0 + 1) * P];
            c = __builtin_amdgcn_wmma_f32_16x16x4_f32(false, a, false, b, (short)0, c,
                                                      false, false);
        }
    }

    const bool colok = (P % 16 == 0) ? true : (col < P);
    if (colok) {
        const int rbase = row0 + ((lane >> 4) << 3);  // rows r / r+8 split on lane halves
        const int ev    = row0 >> 11;                 // row0 / 2048 (tiles never cross events)
#pragma unroll
        for (int r = 0; r < 8; ++r) {
            float v = c[r];
            if constexpr (BIAS_MODE == 1) v += bias[col];
            if constexpr (BIAS_MODE == 2) v += bias[(size_t)ev * P + col];
            if constexpr (ACT == 1) v = tanhf(v);
            D[(size_t)(rbase + r) * P + col] = v;
        }
    }
}

// ---------------------------------------------------------------------------
// kNN: per event (2048 hits, 4-D learned coords), top-40 nearest per query.
// 64 threads/block handle 64 queries; event coords live in LDS (32 KB);
// per-thread 40-entry max-heap in LDS, lane-major (bank-conflict free).
// Results are heap-sorted ascending in d^2 (matches top_k ordering).
// ---------------------------------------------------------------------------
__global__ void __launch_bounds__(64) knn_kernel(const float* __restrict__ s,
                                                 int* __restrict__ nbr_idx,
                                                 float* __restrict__ nbr_w)
{
    __shared__ float4 s_sh[Mv];        // 32 KB
    __shared__ float  hd[Kv * 64];     // 10 KB
    __shared__ int    hi[Kv * 64];     // 10 KB

    const int tid   = threadIdx.x;
    const int b     = blockIdx.x >> 5;   // 32 chunks per event
    const int chunk = blockIdx.x & 31;

    const float4* s4 = (const float4*)s + (size_t)b * Mv;
    for (int i = tid; i < Mv; i += 64) s_sh[i] = s4[i];
#pragma unroll
    for (int k = 0; k < Kv; ++k) { hd[k * 64 + tid] = FLT_MAX; hi[k * 64 + tid] = 0; }
    __syncthreads();

    const int    ql = chunk * 64 + tid;
    const float4 q  = s_sh[ql];

    for (int j = 0; j < Mv; ++j) {
        const float4 p = s_sh[j];
        const float dx = q.x - p.x, dy = q.y - p.y, dz = q.z - p.z, dw = q.w - p.w;
        const float d  = dx * dx + dy * dy + dz * dz + dw * dw;
        if (d < hd[tid]) {              // beats current 40th smallest -> replace root
            int pos = 0;
            for (;;) {
                int ch = 2 * pos + 1;
                if (ch >= Kv) break;
                float dc = hd[ch * 64 + tid];
                const int c2 = ch + 1;
                if (c2 < Kv) { const float dc2 = hd[c2 * 64 + tid]; if (dc2 > dc) { ch = c2; dc = dc2; } }
                if (dc <= d) break;
                hd[pos * 64 + tid] = dc;
                hi[pos * 64 + tid] = hi[ch * 64 + tid];
                pos = ch;
            }
            hd[pos * 64 + tid] = d;
            hi[pos * 64 + tid] = j;
        }
    }

    // in-place heapsort -> ascending distances in hd[0..39]
    for (int e = Kv - 1; e > 0; --e) {
        const float dmax = hd[tid];
        const int   imax = hi[tid];
        float d  = hd[e * 64 + tid];
        int   ji = hi[e * 64 + tid];
        hd[e * 64 + tid] = dmax;
        hi[e * 64 + tid] = imax;
        int pos = 0;
        for (;;) {
            int ch = 2 * pos + 1;
            if (ch >= e) break;
            float dc = hd[ch * 64 + tid];
            const int c2 = ch + 1;
            if (c2 < e) { const float dc2 = hd[c2 * 64 + tid]; if (dc2 > dc) { ch = c2; dc = dc2; } }
            if (dc <= d) break;
            hd[pos * 64 + tid] = dc;
            hi[pos * 64 + tid] = hi[ch * 64 + tid];
            pos = ch;
        }
        hd[pos * 64 + tid] = d;
        hi[pos * 64 + tid] = ji;
    }

    const size_t n = (size_t)b * Mv + ql;
    for (int k = 0; k < Kv; ++k) {
        const float d  = hd[k * 64 + tid];
        const int   gi = b * Mv + hi[k * 64 + tid];   // global hit index
        nbr_idx[n * Kv + k] = gi;
        nbr_w[n * Kv + k]   = expf(-10.f * d);
    }
}

// mean/max over the 40 weighted neighbor messages; aggr = [mean(22) | max(22)]
__global__ void aggr_kernel(const float* __restrict__ h, const int* __restrict__ nbr_idx,
                            const float* __restrict__ nbr_w, float* __restrict__ aggr)
{
    const int t = blockIdx.x * blockDim.x + threadIdx.x;
    if (t >= Nv * PDv) return;
    const int n  = t / PDv;
    const int pd = t - n * PDv;
    float sum = 0.f, mx = -FLT_MAX;
#pragma unroll
    for (int k = 0; k < Kv; ++k) {
        const int   gi = nbr_idx[(size_t)n * Kv + k];
        const float wv = nbr_w[(size_t)n * Kv + k];
        const float v  = h[(size_t)gi * PDv + pd] * wv;
        sum += v;
        mx = fmaxf(mx, v);
    }
    aggr[(size_t)n * (2 * PDv) + pd]       = sum * (1.f / Kv);
    aggr[(size_t)n * (2 * PDv) + PDv + pd] = mx;
}

// per-event mean/min/max of y -> mmm[B, 288] ( [mean(96) | min(96) | max(96)] )
__global__ void mmm_kernel(const float* __restrict__ y, float* __restrict__ mmm)
{
    const int col = threadIdx.x;   // 96
    const int b   = blockIdx.x;    // 16
    float sum = 0.f, mn = FLT_MAX, mx = -FLT_MAX;
    for (int r = 0; r < Mv; ++r) {
        const float v = y[((size_t)b * Mv + r) * OUTv + col];
        sum += v;
        mn = fminf(mn, v);
        mx = fmaxf(mx, v);
    }
    mmm[(size_t)b * 288 + col]       = sum * (1.f / Mv);
    mmm[(size_t)b * 288 + 96 + col]  = mn;
    mmm[(size_t)b * 288 + 192 + col] = mx;
}

// per-event constant: cb[b] = mmm[b] @ W3[0:288] + b3   (folded into final GEMM bias)
__global__ void cb_kernel(const float* __restrict__ mmm, const float* __restrict__ W3,
                          const float* __restrict__ b3, float* __restrict__ cb)
{
    const int col = threadIdx.x;   // 96
    const int b   = blockIdx.x;    // 16
    float acc = b3[col];
    for (int i = 0; i < 288; ++i)
        acc += mmm[(size_t)b * 288 + i] * W3[(size_t)i * 96 + col];
    cb[(size_t)b * 96 + col] = acc;
}

// edge_index [2, N*K] int64: src = global neighbor idx, tgt = repeat(arange(N), K)
__global__ void edge_kernel(const int* __restrict__ nbr_idx, long long* __restrict__ e)
{
    const int t = blockIdx.x * blockDim.x + threadIdx.x;
    if (t >= Nv * Kv) return;
    e[t]                   = (long long)nbr_idx[t];
    e[(size_t)Nv * Kv + t] = (long long)(t / Kv);
}

static inline int gemm_blocks(int P)
{
    const int tilesX = (P + 15) >> 4;
    const int tiles  = (Nv >> 4) * tilesX;
    return (tiles + 7) / 8;   // 8 waves (256 threads) per block
}

extern "C" void kernel_launch(void* const* d_in, const int* in_sizes, int n_in,
                              void* d_out, int out_size, void* d_ws, size_t ws_size,
                              hipStream_t stream)
{
    const float* x    = (const float*)d_in[0];
    /* d_in[1] = batch (unused: events are equal-size, event = row / 2048) */
    const float* W_s  = (const float*)d_in[2];
    const float* b_s  = (const float*)d_in[3];
    const float* W_h  = (const float*)d_in[4];
    const float* b_h  = (const float*)d_in[5];
    const float* W_o1 = (const float*)d_in[6];
    const float* W_o2 = (const float*)d_in[7];
    const float* b_o2 = (const float*)d_in[8];
    const float* W1   = (const float*)d_in[9];
    const float* b1   = (const float*)d_in[10];
    const float* W2   = (const float*)d_in[11];
    const float* b2   = (const float*)d_in[12];
    const float* W3   = (const float*)d_in[13];
    const float* b3   = (const float*)d_in[14];

    // workspace carve-out (256B aligned slices)
    char*  ws  = (char*)d_ws;
    size_t off = 0;
    auto alloc = [&](size_t bytes) -> void* {
        void* p = ws + off;
        off += (bytes + 255) & ~(size_t)255;
        return p;
    };
    float* s_ws   = (float*)alloc((size_t)Nv * SDv * 4);
    float* h_ws   = (float*)alloc((size_t)Nv * PDv * 4);
    int*   idx_ws = (int*)  alloc((size_t)Nv * Kv * 4);
    float* w_ws   = (float*)alloc((size_t)Nv * Kv * 4);
    float* aggr   = (float*)alloc((size_t)Nv * 2 * PDv * 4);
    float* out_ws = (float*)alloc((size_t)Nv * OUTv * 4);
    float* t1_ws  = (float*)alloc((size_t)Nv * 128 * 4);
    float* y_ws   = (float*)alloc((size_t)Nv * OUTv * 4);
    float* mmm_ws = (float*)alloc((size_t)Bv * 288 * 4);
    float* cb_ws  = (float*)alloc((size_t)Bv * 96 * 4);

    float*     z     = (float*)d_out;
    long long* edges = (long long*)((char*)d_out + (size_t)Nv * OUTv * 4);

    // 1-2. s = x@W_s + b_s ; h = x@W_h + b_h
    gemm_wmma_f32<INv, 0, SDv, 1, 0><<<gemm_blocks(SDv), 256, 0, stream>>>(
        x, W_s, nullptr, nullptr, b_s, s_ws, Nv);
    gemm_wmma_f32<INv, 0, PDv, 1, 0><<<gemm_blocks(PDv), 256, 0, stream>>>(
        x, W_h, nullptr, nullptr, b_h, h_ws, Nv);
    // 3. kNN (top-40, ascending d^2) + edge weights
    knn_kernel<<<Bv * 32, 64, 0, stream>>>(s_ws, idx_ws, w_ws);
    // 4. weighted mean/max message aggregation
    {
        const int tot = Nv * PDv;
        aggr_kernel<<<(tot + 255) / 256, 256, 0, stream>>>(h_ws, idx_ws, w_ws, aggr);
    }
    // 5. out = x@W_o1 + aggr@W_o2 + b_o2
    gemm_wmma_f32<INv, 2 * PDv, OUTv, 1, 0><<<gemm_blocks(OUTv), 256, 0, stream>>>(
        x, W_o1, aggr, W_o2, b_o2, out_ws, Nv);
    // 6-7. y = tanh(tanh(out@W1 + b1)@W2 + b2)
    gemm_wmma_f32<OUTv, 0, 128, 1, 1><<<gemm_blocks(128), 256, 0, stream>>>(
        out_ws, W1, nullptr, nullptr, b1, t1_ws, Nv);
    gemm_wmma_f32<128, 0, OUTv, 1, 1><<<gemm_blocks(OUTv), 256, 0, stream>>>(
        t1_ws, W2, nullptr, nullptr, b2, y_ws, Nv);
    // 8-9. per-event mean/min/max, folded top-of-W3 contribution
    mmm_kernel<<<Bv, 96, 0, stream>>>(y_ws, mmm_ws);
    cb_kernel<<<Bv, 96, 0, stream>>>(mmm_ws, W3, b3, cb_ws);
    // 10. z = tanh(y@W3[288:384] + cb[event])
    gemm_wmma_f32<OUTv, 0, OUTv, 2, 1><<<gemm_blocks(OUTv), 256, 0, stream>>>(
        y_ws, W3 + (size_t)288 * 96, nullptr, nullptr, cb_ws, z, Nv);
    // 11. edge_index (int64) after the float z region
    {
        const int tot = Nv * Kv;
        edge_kernel<<<(tot + 255) / 256, 256, 0, stream>>>(idx_ws, edges);
    }
}